// Hitmiss_70497593197434
// MI455X (gfx1250) — compile-verified
//
#include <hip/hip_runtime.h>

#define H   4096
#define W   4096
#define OH  4092
#define OW  4092
#define KS  5
#define BW  128              // output cols per block
#define BH  32               // output rows per block
#define IW  (BW + KS - 1)    // 132 input cols staged
#define IH  (BH + KS - 1)    // 36 input rows staged
#define LP  136              // LDS pitch in floats (132 + 4 pad)
#define QPR (IW / 4)         // 33 float4 chunks per staged row
#define NCH (IH * QPR)       // 1188 chunks total

// Async DMA one float4 from global memory into LDS (CDNA5 async-copy path,
// tracked by ASYNCcnt; bypasses VGPRs entirely).
__device__ __forceinline__ void async_copy_f4(unsigned lds_off, const float* gptr) {
    asm volatile("global_load_async_to_lds_b128 %0, %1, off"
                 :: "v"(lds_off), "v"(gptr)
                 : "memory");
}

__device__ __forceinline__ void wait_async_all() {
    asm volatile("s_wait_asynccnt 0x0" ::: "memory");
}

__global__ __launch_bounds__(256)
void hitmiss_kernel(const float* __restrict__ in,
                    const float* __restrict__ Khit,
                    const float* __restrict__ Kmiss,
                    float* __restrict__ out)
{
    __shared__ float tile[IH * LP];   // 19,584 bytes of 320 KB/WGP

    const int tid  = threadIdx.x;
    const int col0 = blockIdx.x * BW;
    const int row0 = blockIdx.y * BH;

    // ---- Stage the (IH x IW) input tile into LDS with async copies ----
    #pragma unroll
    for (int i = 0; i < 5; ++i) {               // ceil(1188/256) = 5
        int c = tid + i * 256;
        if (c < NCH) {
            int r = c / QPR;
            int q = c - r * QPR;
            int grow = row0 + r;      if (grow > H - 1) grow = H - 1;
            int gcol = col0 + q * 4;  if (gcol > W - 4) gcol = W - 4;
            // low 32 bits of a generic LDS pointer == LDS byte offset
            unsigned loff = (unsigned)(unsigned long long)(uintptr_t)&tile[r * LP + q * 4];
            const float* gptr = in + ((long)grow * W + gcol);
            async_copy_f4(loff, gptr);
        }
    }
    wait_async_all();       // each wave drains its own ASYNCcnt ...
    __syncthreads();        // ... then the barrier makes the whole tile visible

    // ---- Hoist the 50 kernel taps into scalar registers ----
    float kh[25], km[25];
    #pragma unroll
    for (int i = 0; i < 25; ++i) { kh[i] = Khit[i]; km[i] = Kmiss[i]; }

    // ---- Each thread computes a 4x4 output block ----
    const int tx = tid & 31;          // 32 col-groups of 4
    const int ty = tid >> 5;          // 8 row-groups of 4
    const int lc = tx * 4;            // local input col of output col 0
    const int lr = ty * 4;            // local input row of output row 0

    const float INF = __builtin_inff();
    float4 mn[4], mx[4];
    #pragma unroll
    for (int r = 0; r < 4; ++r) {
        mn[r] = make_float4( INF,  INF,  INF,  INF);
        mx[r] = make_float4(-INF, -INF, -INF, -INF);
    }

    #pragma unroll
    for (int rin = 0; rin < 8; ++rin) {
        // one wave reads one LDS row: 32 consecutive float4 -> conflict-free
        const float4* p = (const float4*)&tile[(lr + rin) * LP + lc];
        float4 a = p[0], b = p[1];
        float f[8] = { a.x, a.y, a.z, a.w, b.x, b.y, b.z, b.w };

        #pragma unroll
        for (int ro = 0; ro < 4; ++ro) {
            const int di = rin - ro;             // compile-time after unroll
            if (di >= 0 && di < KS) {
                #pragma unroll
                for (int dj = 0; dj < KS; ++dj) {
                    const float h = kh[di * 5 + dj];
                    const float m = km[di * 5 + dj];
                    // sequential chains -> clang fuses into v_min3/v_max3
                    mn[ro].x = fminf(mn[ro].x, f[dj + 0] - h);
                    mn[ro].y = fminf(mn[ro].y, f[dj + 1] - h);
                    mn[ro].z = fminf(mn[ro].z, f[dj + 2] - h);
                    mn[ro].w = fminf(mn[ro].w, f[dj + 3] - h);
                    mx[ro].x = fmaxf(mx[ro].x, f[dj + 0] - m);
                    mx[ro].y = fmaxf(mx[ro].y, f[dj + 1] - m);
                    mx[ro].z = fmaxf(mx[ro].z, f[dj + 2] - m);
                    mx[ro].w = fmaxf(mx[ro].w, f[dj + 3] - m);
                }
            }
        }
    }

    // ---- Store: full float4, 16B-aligned (OW % 4 == 0) ----
    const int ox = col0 + lc;
    if (ox + 3 < OW) {
        #pragma unroll
        for (int r = 0; r < 4; ++r) {
            const int oy = row0 + lr + r;
            if (oy < OH) {
                float4 v = make_float4(mn[r].x - mx[r].x,
                                       mn[r].y - mx[r].y,
                                       mn[r].z - mx[r].z,
                                       mn[r].w - mx[r].w);
                *(float4*)(out + (long)oy * OW + ox) = v;
            }
        }
    }
}

extern "C" void kernel_launch(void* const* d_in, const int* in_sizes, int n_in,
                              void* d_out, int out_size, void* d_ws, size_t ws_size,
                              hipStream_t stream) {
    const float* in    = (const float*)d_in[0];   // 4096*4096 f32
    const float* Khit  = (const float*)d_in[1];   // 25 f32
    const float* Kmiss = (const float*)d_in[2];   // 25 f32
    float* out = (float*)d_out;                   // 4092*4092 f32

    dim3 block(256);
    dim3 grid((OW + BW - 1) / BW, (OH + BH - 1) / BH);   // 32 x 128
    hitmiss_kernel<<<grid, block, 0, stream>>>(in, Khit, Kmiss, out);
}